// BondOrderInteraction_47425028883061
// MI455X (gfx1250) — compile-verified
//
#include <hip/hip_runtime.h>

typedef __attribute__((ext_vector_type(2))) float v2f;
typedef __attribute__((ext_vector_type(8))) float v8f;

#define F_IN    256
#define CUT_R   3.9f
#define CUT_D   0.1f
#define PI_F    3.14159265358979323846f

__global__ void zero_f32_kernel(float* __restrict__ p, int n) {
    int i = blockIdx.x * blockDim.x + threadIdx.x;
    if (i < n) p[i] = 0.0f;
}

// Fused per-node projection using fp32 WMMA (V_WMMA_F32_16X16X4_F32).
// e_buf[node][0..3] = feat @ W_src + b_src ; e_buf[node][4..7] = feat @ W_dst
// One wave handles a 16-node tile: M = node row (lane&15), N = output col,
// K stepped by 4 per WMMA (64 WMMAs over F_IN=256), fp32-exact accumulate.
__global__ __launch_bounds__(256) void node_proj_wmma_kernel(
    const float* __restrict__ feat,
    const float* __restrict__ W_src,
    const float* __restrict__ b_src,
    const float* __restrict__ W_dst,
    float* __restrict__ e_buf,
    int n_nodes, int n_tiles)
{
    // Stage combined B matrix [K=256][N=16] in LDS, cols 8..15 zero padded
    // so every lane can load unconditionally (keeps EXEC uniform around WMMA).
    __shared__ float Bs[F_IN * 16];
    for (int idx = threadIdx.x; idx < F_IN * 16; idx += blockDim.x) {
        int k = idx >> 4, n = idx & 15;
        float v = 0.0f;
        if (n < 4)      v = W_src[k * 4 + n];
        else if (n < 8) v = W_dst[k * 4 + (n - 4)];
        Bs[idx] = v;
    }
    __syncthreads();

    const int wave = threadIdx.x >> 5;
    const int lane = threadIdx.x & 31;
    const int tile = blockIdx.x * (blockDim.x >> 5) + wave;
    if (tile >= n_tiles) return;   // wave-uniform exit: surviving waves keep EXEC all-1s

    const int n    = lane & 15;    // A: row M index / B: col N index
    const int half = lane >> 4;    // selects K sub-pair per the 32-bit A layout

    int arow = tile * 16 + n;
    if (arow >= n_nodes) arow = n_nodes - 1;   // safe duplicate load (store is guarded)
    const float* frow = feat + (long)arow * F_IN + 2 * half;

    v8f acc = {};
#pragma unroll 8
    for (int k0 = 0; k0 < F_IN; k0 += 4) {
        const int kk = k0 + 2 * half;          // lanes 0-15: K=k0,k0+1 ; lanes 16-31: K=k0+2,k0+3
        float2 av = *reinterpret_cast<const float2*>(frow + k0);  // 8B-aligned (kk even)
        v2f a; a.x = av.x; a.y = av.y;
        v2f b; b.x = Bs[kk * 16 + n]; b.y = Bs[(kk + 1) * 16 + n];
        // 8 args: (neg_a, A, neg_b, B, c_mod, C, reuse_a, reuse_b)
        acc = __builtin_amdgcn_wmma_f32_16x16x4_f32(
            false, a, false, b, (short)0, acc, false, false);
    }

    // C/D layout: VGPR r, lanes 0-15 -> M=r ; lanes 16-31 -> M=r+8 ; N=lane&15
    const float bb = (n < 4) ? b_src[n] : 0.0f;   // bias only on the e_src columns
    if (n < 8) {
        const int mbase = tile * 16 + 8 * half;
#pragma unroll
        for (int r = 0; r < 8; ++r) {
            const int node = mbase + r;
            if (node < n_nodes) e_buf[(long)node * 8 + n] = acc[r] + bb;
        }
    }
}

// One thread per edge: gather e-vectors, fused exp algebra, cutoff, atomic scatter.
__global__ __launch_bounds__(256) void edge_scatter_kernel(
    const float* __restrict__ e_buf,
    const float* __restrict__ bond_order,
    const float* __restrict__ bondlength,
    const int*  __restrict__ src,
    const int*  __restrict__ dst,
    float* __restrict__ out,
    int n_edges)
{
    int i = blockIdx.x * blockDim.x + threadIdx.x;
    if (i >= n_edges) return;

    const float r = bondlength[i];
    if (r > CUT_R + CUT_D) return;             // cutoff == 0 -> contributes nothing

    const int s = src[i];
    const int d = dst[i];
    const float4 es = *reinterpret_cast<const float4*>(e_buf + (long)s * 8);      // e_src part
    const float4 ed = *reinterpret_cast<const float4*>(e_buf + (long)d * 8 + 4);  // e_dst part

    const float e0 = es.x + ed.x;
    const float e1 = es.y + ed.y;
    const float e2 = es.z + ed.z;
    const float e3 = es.w + ed.w;

    // params[k] = exp(ek);  p0*exp(-p1*r) == exp(e0 - exp(e1)*r)
    const float f_rep = __expf(e0 - __expf(e1) * r);
    const float f_att = __expf(e2 - __expf(e3) * r);

    float c = 1.0f;
    if (r >= CUT_R - CUT_D)
        c = 0.5f - 0.5f * __sinf(PI_F * (r - CUT_R) / (2.0f * CUT_D));

    const float V = c * (f_rep - bond_order[i] * f_att);
    unsafeAtomicAdd(&out[d], V);               // hardware global_atomic_add_f32
}

extern "C" void kernel_launch(void* const* d_in, const int* in_sizes, int n_in,
                              void* d_out, int out_size, void* d_ws, size_t ws_size,
                              hipStream_t stream)
{
    const float* feat       = (const float*)d_in[0];
    const float* bond_order = (const float*)d_in[1];
    const float* bondlength = (const float*)d_in[2];
    const int*   src        = (const int*)d_in[3];
    const int*   dst        = (const int*)d_in[4];
    const float* W_src      = (const float*)d_in[5];
    const float* b_src      = (const float*)d_in[6];
    const float* W_dst      = (const float*)d_in[7];

    float* out   = (float*)d_out;
    float* e_buf = (float*)d_ws;               // n_nodes * 8 floats (3.2 MB)

    const int n_nodes = in_sizes[0] / F_IN;
    const int n_edges = in_sizes[1];
    const int n_tiles = (n_nodes + 15) / 16;

    // 1) zero the output accumulator
    zero_f32_kernel<<<(out_size + 255) / 256, 256, 0, stream>>>(out, out_size);

    // 2) fused dual-projection via WMMA; 8 waves (tiles) per 256-thread block
    const int blocks_proj = (n_tiles + 7) / 8;
    node_proj_wmma_kernel<<<blocks_proj, 256, 0, stream>>>(
        feat, W_src, b_src, W_dst, e_buf, n_nodes, n_tiles);

    // 3) per-edge gather/compute/atomic-scatter
    edge_scatter_kernel<<<(n_edges + 255) / 256, 256, 0, stream>>>(
        e_buf, bond_order, bondlength, src, dst, out, n_edges);
}